// LatentDecoder_50431505990243
// MI455X (gfx1250) — compile-verified
//
#include <hip/hip_runtime.h>

typedef __attribute__((ext_vector_type(2))) float v2f;
typedef __attribute__((ext_vector_type(8))) float v8f;

#define CDIM 4096   // channels (K of big GEMM, also M=output channels)
#define HWDIM 4096  // 64*64 spatial positions (N of big GEMM)
#define BM 64
#define BN 128
#define BK 32
#define A_STRIDE (BK + 4)    // pad: column reads hit 16 distinct banks; row stride 144B (16B-aligned)
#define BP_STRIDE 288        // floats per K-pair row of B (256 data + 32 pad -> hi/lo lane
                             // groups land on disjoint bank halves for b64 reads)

// ---------------------------------------------------------------------------
// Kernel 1: planeT[hw][o] = sum_c w_up[o][c] * bilinear_up(latent)[c][hw] + b_up[o]
// A tile staged via GLOBAL_LOAD_ASYNC_TO_LDS_B128 (ASYNCcnt), overlapped with
// the VALU bilinear construction of the B tile. WMMA f32 16x16x4 compute.
// B tile stored K-pair interleaved: Bs[(k>>1)*288 + n*2 + (k&1)] so a B
// fragment is one 8B-aligned ds_load_b64 (no cross-lane packing movs).
// ---------------------------------------------------------------------------
__global__ __launch_bounds__(256) void fused_upsample_gemm(
    const float* __restrict__ latent,   // (4096, 32, 32)
    const float* __restrict__ w_up,     // (4096, 4096) [o][c]
    const float* __restrict__ b_up,     // (4096)
    float* __restrict__ planeT)         // (4096 hw, 4096 o)
{
    __shared__ float As[BM][A_STRIDE];            // [m][k]
    __shared__ float Bs[(BK / 2) * BP_STRIDE];    // [k/2][n][2], padded

    const int tid  = threadIdx.x;
    const int wave = tid >> 5;          // 0..7
    const int lane = tid & 31;
    const int wm   = wave >> 2;         // 0..1  (M dir)
    const int wn   = wave & 3;          // 0..3  (N dir)
    const int l16  = lane & 15;
    const int hi   = lane >> 4;         // 0/1 -> K sub-pair per ISA layout

    const int m_base = blockIdx.x * BM;   // gridDim.x = 64
    const int n_base = blockIdx.y * BN;   // gridDim.y = 32

    v8f acc[2][2];
    const v8f vzero = {0.f,0.f,0.f,0.f,0.f,0.f,0.f,0.f};
    #pragma unroll
    for (int i = 0; i < 2; ++i)
        #pragma unroll
        for (int j = 0; j < 2; ++j) acc[i][j] = vzero;

    // per-thread A-tile slots (2 x b128 per thread), invariant across kt
    const int a_idx0 = tid;              // 0..255
    const int a_idx1 = 256 + tid;        // 256..511
    const int a_row0 = a_idx0 >> 3, a_c40 = (a_idx0 & 7) << 2;
    const int a_row1 = a_idx1 >> 3, a_c41 = (a_idx1 & 7) << 2;
    const unsigned lds_a0 = (unsigned)(uintptr_t)&As[a_row0][a_c40];
    const unsigned lds_a1 = (unsigned)(uintptr_t)&As[a_row1][a_c41];

    for (int kt = 0; kt < CDIM; kt += BK) {
        // ---- A tile: async copy w_up[m_base+row][kt+col] -> LDS (16B per lane)
        {
            unsigned long long ga0 =
                (unsigned long long)(w_up + (size_t)(m_base + a_row0) * CDIM + kt + a_c40);
            unsigned long long ga1 =
                (unsigned long long)(w_up + (size_t)(m_base + a_row1) * CDIM + kt + a_c41);
            asm volatile("global_load_async_to_lds_b128 %0, %1, off"
                         :: "v"(lds_a0), "v"(ga0) : "memory");
            asm volatile("global_load_async_to_lds_b128 %0, %1, off"
                         :: "v"(lds_a1), "v"(ga1) : "memory");
        }

        // ---- B tile: up[kt+r][n_base+col] computed on the fly from latent
        //      (overlaps with the in-flight async A copy)
        #pragma unroll
        for (int i = 0; i < 16; ++i) {
            int idx = i * 256 + tid;            // 0..4095
            int r   = idx >> 7;                 // 0..31 (k within tile)
            int col = idx & 127;
            int hw  = n_base + col;
            int h   = hw >> 6;
            int w   = hw & 63;
            // half-pixel src coords for 2x upsample
            float cy = h * 0.5f - 0.25f;
            float cx = w * 0.5f - 0.25f;
            float y0f = floorf(cy), x0f = floorf(cx);
            float fy = cy - y0f, fx = cx - x0f;
            int y0 = (int)y0f, x0 = (int)x0f;
            int y0c = min(max(y0, 0), 31), y1c = min(max(y0 + 1, 0), 31);
            int x0c = min(max(x0, 0), 31), x1c = min(max(x0 + 1, 0), 31);
            const float* Lc = latent + (size_t)(kt + r) * 1024;
            float v00 = Lc[y0c * 32 + x0c];
            float v01 = Lc[y0c * 32 + x1c];
            float v10 = Lc[y1c * 32 + x0c];
            float v11 = Lc[y1c * 32 + x1c];
            float vt = v00 + fx * (v01 - v00);
            float vb = v10 + fx * (v11 - v10);
            Bs[(r >> 1) * BP_STRIDE + col * 2 + (r & 1)] = vt + fy * (vb - vt);
        }

        // drain this wave's async copies, then make tiles visible block-wide
        asm volatile("s_wait_asynccnt 0x0" ::: "memory");
        __syncthreads();

        // ---- compute: 8 k-steps x 4 wmma
        #pragma unroll
        for (int k0 = 0; k0 < BK; k0 += 4) {
            v2f a[2], b[2];
            #pragma unroll
            for (int i = 0; i < 2; ++i) {
                int m = wm * 32 + i * 16 + l16;
                a[i][0] = As[m][k0 + hi * 2 + 0];   // ISA: VGPR0 = K0 (lo lanes) / K2 (hi)
                a[i][1] = As[m][k0 + hi * 2 + 1];   //      VGPR1 = K1 / K3
            }
            #pragma unroll
            for (int j = 0; j < 2; ++j) {
                int n = wn * 32 + j * 16 + l16;
                // K-pair (k0+2*hi, k0+2*hi+1) contiguous -> single b64 load
                b[j] = *(const v2f*)&Bs[((k0 >> 1) + hi) * BP_STRIDE + n * 2];
            }
            #pragma unroll
            for (int i = 0; i < 2; ++i)
                #pragma unroll
                for (int j = 0; j < 2; ++j)
                    acc[i][j] = __builtin_amdgcn_wmma_f32_16x16x4_f32(
                        false, a[i], false, b[j], (short)0, acc[i][j], false, false);
        }
        __syncthreads();
    }

    // ---- epilogue: + bias, store transposed planeT[n][m]
    #pragma unroll
    for (int i = 0; i < 2; ++i) {
        #pragma unroll
        for (int j = 0; j < 2; ++j) {
            int mg0 = m_base + wm * 32 + i * 16 + hi * 8;  // VGPR r -> row mg0+r
            int ng  = n_base + wn * 32 + j * 16 + l16;
            float* dst = planeT + (size_t)ng * CDIM + mg0;
            float4 lo4, hi4;
            lo4.x = acc[i][j][0] + b_up[mg0 + 0];
            lo4.y = acc[i][j][1] + b_up[mg0 + 1];
            lo4.z = acc[i][j][2] + b_up[mg0 + 2];
            lo4.w = acc[i][j][3] + b_up[mg0 + 3];
            hi4.x = acc[i][j][4] + b_up[mg0 + 4];
            hi4.y = acc[i][j][5] + b_up[mg0 + 5];
            hi4.z = acc[i][j][6] + b_up[mg0 + 6];
            hi4.w = acc[i][j][7] + b_up[mg0 + 7];
            *(float4*)(dst)     = lo4;
            *(float4*)(dst + 4) = hi4;
        }
    }
}

// ---------------------------------------------------------------------------
// Kernel 2: one block per unique (z,x) pair (4096 blocks).
// Bicubic 16-tap gather over planeT -> g[4096] in LDS, then per-y lerp + MLP.
// ---------------------------------------------------------------------------
__device__ __forceinline__ float cubic_k(float x) {
    const float A = -0.75f;
    x = fabsf(x);
    float x2 = x * x, x3 = x2 * x;
    float w_in  = (A + 2.f) * x3 - (A + 3.f) * x2 + 1.f;
    float w_out = A * x3 - 5.f * A * x2 + 8.f * A * x - 4.f * A;
    return x <= 1.f ? w_in : (x < 2.f ? w_out : 0.f);
}

__global__ __launch_bounds__(64) void sample_mlp(
    const float* __restrict__ planeT,   // (4096 hw, 4096 c)
    const float* __restrict__ W1, const float* __restrict__ b1,   // (64,128),(128)
    const float* __restrict__ W2, const float* __restrict__ b2,   // (128,128),(128)
    const float* __restrict__ W3, const float* __restrict__ b3,   // (128,1),(1)
    float* __restrict__ out)            // (64,64,64,1): ((iz*64)+iy)*64+ix
{
    __shared__ float g[4096];           // 16 KB
    __shared__ float h1s[128 * 64];     // 32 KB, [j][tid]: lane==bank, conflict-free

    const int tid  = threadIdx.x;       // 0..63 (one y-sample per thread)
    const int pair = blockIdx.x;        // 0..4095
    const int iz = pair >> 6;
    const int ix = pair & 63;

    // bicubic taps (uniform across block)
    float xq = -1.f + 2.f * ix / 63.f;
    float zq = -1.f + 2.f * iz / 63.f;
    float fx = (xq + 1.f) * 0.5f * 63.f;
    float fz = (zq + 1.f) * 0.5f * 63.f;
    float x0f = floorf(fx), z0f = floorf(fz);
    float tx = fx - x0f, tz = fz - z0f;
    float wx[4], wz[4];
    wx[0] = cubic_k(1.f + tx); wx[1] = cubic_k(tx);
    wx[2] = cubic_k(1.f - tx); wx[3] = cubic_k(2.f - tx);
    wz[0] = cubic_k(1.f + tz); wz[1] = cubic_k(tz);
    wz[2] = cubic_k(1.f - tz); wz[3] = cubic_k(2.f - tz);
    int xi[4], zi[4];
    #pragma unroll
    for (int i = 0; i < 4; ++i) {
        xi[i] = min(max((int)x0f + i - 1, 0), 63);
        zi[i] = min(max((int)z0f + i - 1, 0), 63);
    }

    // gather g[c] = sum_ij wz[i]*wx[j] * planeT[(zi*64+xi)][c]  (coalesced in c)
    for (int c = tid; c < 4096; c += 64) {
        float accg = 0.f;
        #pragma unroll
        for (int i = 0; i < 4; ++i) {
            #pragma unroll
            for (int j = 0; j < 4; ++j) {
                float wv = wz[i] * wx[j];
                accg += wv * planeT[(size_t)(zi[i] * 64 + xi[j]) * CDIM + c];
            }
        }
        g[c] = accg;
    }
    __syncthreads();

    // monoplanar lerp along y (samples axis): f[k] in registers
    const int iy = tid;
    float yv = -1.f + 2.f * iy / 63.f;
    float u  = fminf(fmaxf(yv, -1.f), 1.f);
    float s  = (u + 1.f) * 0.5f * 63.f;
    int s0   = min(max((int)floorf(s), 0), 62);
    float t  = s - (float)s0;

    float f[64];
    #pragma unroll
    for (int k = 0; k < 64; ++k) {
        float g0 = g[s0 * 64 + k];
        float g1 = g[s0 * 64 + 64 + k];
        f[k] = g0 + t * (g1 - g0);
    }

    // layer 1: h1 = relu(f @ W1 + b1), staged in LDS per-thread column
    for (int j = 0; j < 128; ++j) {
        float accv = b1[j];
        #pragma unroll
        for (int k = 0; k < 64; ++k)
            accv = fmaf(f[k], W1[k * 128 + j], accv);
        h1s[j * 64 + tid] = fmaxf(accv, 0.f);
    }

    // layers 2+3 fused: out = relu(h1 @ W2 + b2) @ W3 + b3
    float o = b3[0];
    for (int j2 = 0; j2 < 128; ++j2) {
        float accv = b2[j2];
        for (int j1 = 0; j1 < 128; ++j1)
            accv = fmaf(h1s[j1 * 64 + tid], W2[j1 * 128 + j2], accv);
        o = fmaf(fmaxf(accv, 0.f), W3[j2], o);
    }

    out[((iz * 64) + iy) * 64 + ix] = o;
}

// ---------------------------------------------------------------------------
extern "C" void kernel_launch(void* const* d_in, const int* in_sizes, int n_in,
                              void* d_out, int out_size, void* d_ws, size_t ws_size,
                              hipStream_t stream) {
    const float* latent = (const float*)d_in[0];
    const float* w_up   = (const float*)d_in[1];
    const float* b_up   = (const float*)d_in[2];
    const float* W1     = (const float*)d_in[3];
    const float* b1     = (const float*)d_in[4];
    const float* W2     = (const float*)d_in[5];
    const float* b2     = (const float*)d_in[6];
    const float* W3     = (const float*)d_in[7];
    const float* b3     = (const float*)d_in[8];
    // d_in[9..11] = resx/resy/resz = 64 (hardcoded)

    float* planeT = (float*)d_ws;   // 4096*4096 floats = 64 MB scratch

    fused_upsample_gemm<<<dim3(CDIM / BM, HWDIM / BN), 256, 0, stream>>>(
        latent, w_up, b_up, planeT);

    sample_mlp<<<4096, 64, 0, stream>>>(planeT, W1, b1, W2, b2, W3, b3,
                                        (float*)d_out);
}